// Recognizer_3633542332469
// MI455X (gfx1250) — compile-verified
//
#include <hip/hip_runtime.h>
#include <hip/hip_bf16.h>
#include <math.h>

typedef __attribute__((ext_vector_type(2))) float v2f;
typedef __attribute__((ext_vector_type(8))) float v8f;

#define NEGV (-1e30f)

// ---------------------------------------------------------------------------
// Kernel A: logits = feat @ W^T + bias, then log_softmax over V, fp32 WMMA.
// feat: [M, F], W: [V, F], lp out: [M, V].
// Block = 256 threads = 8 waves. Waves tile 2(M-groups of 64) x 4(N-groups of 64).
// Each wave: 4x4 tiles of 16x16 accumulated with V_WMMA_F32_16X16X4_F32.
// ---------------------------------------------------------------------------
__global__ __launch_bounds__(256) void gemm_logsoftmax_kernel(
    const float* __restrict__ feat,
    const float* __restrict__ Wm,
    const float* __restrict__ bias,
    float* __restrict__ lp,
    int M, int F, int V)
{
    const int tid  = threadIdx.x;
    const int wave = tid >> 5;         // 0..7
    const int lane = tid & 31;
    const int lr   = lane & 15;        // row/col within 16-tile
    const int half = lane >> 4;        // lane-half selects K+2 (A/B f32 layout)
    const int nw   = wave >> 1;        // N-group 0..3 (cols n0 = nw*64)
    const int mW   = (wave & 1) * 64;  // M-offset within 128-row block tile
    const int n0   = nw * 64;
    const int koff = half * 2;

    const int mBase = blockIdx.x * 128;

    v8f acc[4][4];
#pragma unroll
    for (int mt = 0; mt < 4; ++mt)
#pragma unroll
        for (int nt = 0; nt < 4; ++nt)
            acc[mt][nt] = (v8f){0.f,0.f,0.f,0.f,0.f,0.f,0.f,0.f};

    // Per-lane base pointers; A layout (16x4 f32): lanes0-15 hold K=k,k+1,
    // lanes16-31 hold K=k+2,k+3 for row M=lane&15. B (4x16) mirrored.
    const float* fb[4];
    const float* wb[4];
#pragma unroll
    for (int mt = 0; mt < 4; ++mt) {
        int m = mBase + mW + mt * 16 + lr;
        if (m > M - 1) m = M - 1;                  // clamp (M % 128 == 0 normally)
        fb[mt] = feat + (size_t)m * F + koff;
    }
#pragma unroll
    for (int nt = 0; nt < 4; ++nt) {
        int n = n0 + nt * 16 + lr;
        wb[nt] = Wm + (size_t)n * F + koff;
    }

    for (int k = 0; k < F; k += 4) {
        v2f a[4], bm[4];
#pragma unroll
        for (int mt = 0; mt < 4; ++mt) a[mt]  = *(const v2f*)(fb[mt] + k);
#pragma unroll
        for (int nt = 0; nt < 4; ++nt) bm[nt] = *(const v2f*)(wb[nt] + k);
#pragma unroll
        for (int mt = 0; mt < 4; ++mt)
#pragma unroll
            for (int nt = 0; nt < 4; ++nt)
                acc[mt][nt] = __builtin_amdgcn_wmma_f32_16x16x4_f32(
                    false, a[mt], false, bm[nt], (short)0, acc[mt][nt],
                    false, false);
    }

    // ---- epilogue: + bias, log_softmax over the V (=256) dimension ----
    float bv[4];
#pragma unroll
    for (int nt = 0; nt < 4; ++nt) bv[nt] = bias[n0 + nt * 16 + lr];
#pragma unroll
    for (int mt = 0; mt < 4; ++mt)
#pragma unroll
        for (int nt = 0; nt < 4; ++nt)
#pragma unroll
            for (int e = 0; e < 8; ++e)
                acc[mt][nt][e] += bv[nt];

    // C/D layout: VGPR e -> row mt*16 + e + half*8, col = n0 + nt*16 + lr.
    __shared__ float red_max[128][4];
    __shared__ float red_sum[128][4];
    __shared__ float lse[128];

#pragma unroll
    for (int mt = 0; mt < 4; ++mt) {
#pragma unroll
        for (int e = 0; e < 8; ++e) {
            float mx = NEGV;
#pragma unroll
            for (int nt = 0; nt < 4; ++nt) mx = fmaxf(mx, acc[mt][nt][e]);
#pragma unroll
            for (int msk = 1; msk < 16; msk <<= 1)
                mx = fmaxf(mx, __shfl_xor(mx, msk, 32));
            float s = 0.f;
#pragma unroll
            for (int nt = 0; nt < 4; ++nt) s += expf(acc[mt][nt][e] - mx);
#pragma unroll
            for (int msk = 1; msk < 16; msk <<= 1)
                s += __shfl_xor(s, msk, 32);
            int row = mW + mt * 16 + half * 8 + e;   // 0..127, per N-group
            if (lr == 0) { red_max[row][nw] = mx; red_sum[row][nw] = s; }
        }
    }
    __syncthreads();
    if (tid < 128) {
        float mx = NEGV;
#pragma unroll
        for (int i = 0; i < 4; ++i) mx = fmaxf(mx, red_max[tid][i]);
        float s = 0.f;
#pragma unroll
        for (int i = 0; i < 4; ++i) s += red_sum[tid][i] * expf(red_max[tid][i] - mx);
        lse[tid] = mx + logf(s);
    }
    __syncthreads();

#pragma unroll
    for (int mt = 0; mt < 4; ++mt) {
#pragma unroll
        for (int e = 0; e < 8; ++e) {
            int row  = mW + mt * 16 + half * 8 + e;
            int grow = mBase + row;
            if (grow < M) {
                float l = lse[row];
                size_t base = (size_t)grow * V + n0 + lr;
#pragma unroll
                for (int nt = 0; nt < 4; ++nt)
                    lp[base + nt * 16] = acc[mt][nt][e] - l;
            }
        }
    }
}

// ---------------------------------------------------------------------------
// Kernel B: CTC forward alpha recursion, one workgroup per batch element.
// Thread tid owns extended-label position l = tid; alpha kept in a REGISTER.
// alpha[l-1], alpha[l-2] via __shfl_up within the wave; wave-boundary values
// (lanes 30,31 of the previous wave) via a tiny parity double-buffered LDS
// edge array -> exactly ONE __syncthreads() per time step. The lp gather for
// step t+1 is issued during step t so its latency hides under the barrier.
// ---------------------------------------------------------------------------
__global__ __launch_bounds__(544) void ctc_alpha_kernel(
    const float* __restrict__ lp,
    const int* __restrict__ targets,
    const int* __restrict__ in_len,
    const int* __restrict__ tg_len,
    float* __restrict__ partial,
    int T, int V, int S)
{
    const int b    = blockIdx.x;
    const int L    = 2 * S + 1;
    const int tid  = threadIdx.x;
    const int wave = tid >> 5;
    const int lane = tid & 31;
    const int NW   = blockDim.x >> 5;

    extern __shared__ float smem[];
    float* afin = smem;              // [L]   final alpha collection
    float* edge = smem + L;          // [2][NW][2] parity-buffered wave edges

    // extended label for this l, and whether the l-2 skip transition is legal
    int e = 0, skip = 0;
    if (tid < L) {
        if (tid & 1) e = targets[b * S + (tid >> 1)];
        int ep = -1;
        if (tid >= 2) ep = (tid & 1) ? targets[b * S + ((tid - 2) >> 1)] : 0;
        skip = (e != 0) && (e != ep);
    }

    const float* lpb = lp + (size_t)b * T * V;
    const int len = in_len[b];

    // ---- t = 0 init (alpha in register) ----
    float a = NEGV;
    if (tid == 0)      a = lpb[0];      // blank
    else if (tid == 1) a = lpb[e];      // first label
    // publish t=0 edges at parity 0
    if (lane >= 30) edge[(0 * NW + wave) * 2 + (lane - 30)] = a;

    // software-pipelined gather of lp[t, e]
    float lpe_cur = (T > 1) ? lpb[(size_t)V + e] : 0.f;
    __syncthreads();

    for (int t = 1; t < T; ++t) {
        // issue next row's gather early; wait lands next iteration
        float lpe_next = 0.f;
        if (t + 1 < T) lpe_next = lpb[(size_t)(t + 1) * V + e];

        const int rp = (t - 1) & 1;   // parity of edges holding alpha(t-1)
        const int wp = t & 1;         // parity we publish alpha(t) into

        float a1 = __shfl_up(a, 1, 32);
        float a2 = __shfl_up(a, 2, 32);
        if (lane == 0) {
            a1 = (wave > 0) ? edge[(rp * NW + (wave - 1)) * 2 + 1] : NEGV;
            a2 = (wave > 0) ? edge[(rp * NW + (wave - 1)) * 2 + 0] : NEGV;
        } else if (lane == 1) {
            a2 = (wave > 0) ? edge[(rp * NW + (wave - 1)) * 2 + 1] : NEGV;
        }
        if (tid < 1) a1 = NEGV;
        if (tid < 2 || !skip) a2 = NEGV;

        float m  = fmaxf(a, fmaxf(a1, a2));
        float s  = expf(a - m) + expf(a1 - m) + expf(a2 - m);
        float nv = lpe_cur + m + logf(s);
        a = (t < len) ? nv : a;

        if (lane >= 30) edge[(wp * NW + wave) * 2 + (lane - 30)] = a;
        __syncthreads();
        lpe_cur = lpe_next;
    }

    // collect final alpha, compute per-sequence loss
    if (tid < L) afin[tid] = a;
    __syncthreads();
    if (tid == 0) {
        int   tl = tg_len[b];
        float x  = afin[2 * tl];
        float y  = afin[2 * tl - 1];
        float m  = fmaxf(x, y);
        float per = -(m + logf(expf(x - m) + expf(y - m)));
        partial[b] = per / (float)tl;
    }
}

// ---------------------------------------------------------------------------
// Kernel C: mean over batch -> scalar output.
// ---------------------------------------------------------------------------
__global__ __launch_bounds__(32) void reduce_mean_kernel(
    const float* __restrict__ partial, float* __restrict__ out, int B)
{
    float v = (threadIdx.x < B) ? partial[threadIdx.x] : 0.f;
#pragma unroll
    for (int msk = 16; msk >= 1; msk >>= 1) v += __shfl_xor(v, msk, 32);
    if (threadIdx.x == 0) out[0] = v / (float)B;
}

// ---------------------------------------------------------------------------
extern "C" void kernel_launch(void* const* d_in, const int* in_sizes, int n_in,
                              void* d_out, int out_size, void* d_ws, size_t ws_size,
                              hipStream_t stream) {
    const float* feat    = (const float*)d_in[0];  // [B, T, F]
    const float* Wm      = (const float*)d_in[1];  // [V, F]
    const float* bias    = (const float*)d_in[2];  // [V]
    const int*   targets = (const int*)d_in[3];    // [B, S]
    const int*   in_len  = (const int*)d_in[4];    // [B]
    const int*   tg_len  = (const int*)d_in[5];    // [B]

    const int B = in_sizes[4];
    const int V = in_sizes[2];
    const int F = in_sizes[1] / V;
    const int T = in_sizes[0] / (B * F);
    const int S = in_sizes[3] / B;
    const int M = B * T;
    const int L = 2 * S + 1;

    float* lp      = (float*)d_ws;                 // [M, V]
    float* partial = lp + (size_t)M * V;           // [B]

    // 1) fused classifier + log_softmax (fp32 WMMA)
    dim3 gridA((M + 127) / 128);
    gemm_logsoftmax_kernel<<<gridA, 256, 0, stream>>>(feat, Wm, bias, lp, M, F, V);

    // 2) CTC alpha scan, one block per batch element (register alpha + edges)
    int blockB = ((L + 31) / 32) * 32;             // 544 for S=256
    int NW = blockB / 32;
    size_t shmem = (size_t)(L + 4 * NW) * sizeof(float);
    ctc_alpha_kernel<<<B, blockB, shmem, stream>>>(lp, targets, in_len, tg_len,
                                                   partial, T, V, S);

    // 3) batch mean
    reduce_mean_kernel<<<1, 32, 0, stream>>>(partial, (float*)d_out, B);
}